// QuantumEstimatorHead_65481071404354
// MI455X (gfx1250) — compile-verified
//
#include <hip/hip_runtime.h>

// out[i] = sin(x[i]) * sin(w[0])  — pure streaming elementwise op.
//
// Roofline: 268 MB mandatory traffic @ 23.3 TB/s => ~11.5 us floor.
// ~6 VALU ops/element is negligible at that time budget, so the kernel is
// built purely around data movement:
//   - b128 loads (RT policy: input stays resident in the 192 MB L2 across
//     graph replays)
//   - b128 non-temporal stores (write-once output must not evict the input)
//   - global_prefetch_b8 one tile ahead for the cold pass
//   - hardware v_sin_f32 via __sinf (inputs are N(0,1), well in range)
// WMMA is deliberately absent: there is no contraction dimension and no
// matrix op can reduce the mandatory HBM traffic of a 0.25 FLOP/byte kernel.

// Native clang vector (NOT HIP_vector_type) so nontemporal builtins accept it.
typedef __attribute__((ext_vector_type(4))) float v4f;

namespace {

constexpr int BLOCK = 256;          // 8 wave32s
constexpr int V4_PER_THREAD = 4;    // 16 floats / thread, 4 b128 loads in flight

__device__ __forceinline__ v4f sin_mul4(v4f v, float s) {
  v4f r;
  r.x = __sinf(v.x) * s;
  r.y = __sinf(v.y) * s;
  r.z = __sinf(v.z) * s;
  r.w = __sinf(v.w) * s;
  return r;
}

__global__ __launch_bounds__(BLOCK) void qhead_main(
    const v4f* __restrict__ x4, const float* __restrict__ w,
    v4f* __restrict__ o4, int n4) {
  const float sw = __sinf(w[0]);           // uniform: s_load + one v_sin
  const int stride = BLOCK;
  const int base = blockIdx.x * (BLOCK * V4_PER_THREAD) + threadIdx.x;

  // Speculative prefetch one block-tile ahead (gfx1250 global_prefetch_b8).
  // Invalid addresses past the end are silently dropped by hardware.
  __builtin_prefetch(
      reinterpret_cast<const char*>(x4 + base + BLOCK * V4_PER_THREAD), 0, 1);

  const int i0 = base;
  const int i1 = base + stride;
  const int i2 = base + 2 * stride;
  const int i3 = base + 3 * stride;

  if (i3 < n4) {
    // Fast path: 4 independent b128 loads issued back-to-back (ILP),
    // then compute + NT b128 stores.
    v4f v0 = x4[i0];
    v4f v1 = x4[i1];
    v4f v2 = x4[i2];
    v4f v3 = x4[i3];
    __builtin_nontemporal_store(sin_mul4(v0, sw), o4 + i0);
    __builtin_nontemporal_store(sin_mul4(v1, sw), o4 + i1);
    __builtin_nontemporal_store(sin_mul4(v2, sw), o4 + i2);
    __builtin_nontemporal_store(sin_mul4(v3, sw), o4 + i3);
  } else {
    if (i0 < n4) __builtin_nontemporal_store(sin_mul4(x4[i0], sw), o4 + i0);
    if (i1 < n4) __builtin_nontemporal_store(sin_mul4(x4[i1], sw), o4 + i1);
    if (i2 < n4) __builtin_nontemporal_store(sin_mul4(x4[i2], sw), o4 + i2);
    if (i3 < n4) __builtin_nontemporal_store(sin_mul4(x4[i3], sw), o4 + i3);
  }
}

// Scalar tail for n % 4 != 0 (not hit for BATCH = 2^25, kept for generality).
__global__ void qhead_tail(const float* __restrict__ x,
                           const float* __restrict__ w,
                           float* __restrict__ out, int start, int n) {
  int i = start + blockIdx.x * blockDim.x + threadIdx.x;
  if (i < n) out[i] = __sinf(x[i]) * __sinf(w[0]);
}

}  // namespace

extern "C" void kernel_launch(void* const* d_in, const int* in_sizes, int n_in,
                              void* d_out, int out_size, void* d_ws,
                              size_t ws_size, hipStream_t stream) {
  const float* x = (const float*)d_in[0];     // (BATCH, 1) fp32
  const float* w = (const float*)d_in[1];     // (1,) fp32
  float* out = (float*)d_out;                 // (BATCH,) fp32
  const int n = in_sizes[0];
  const int n4 = n >> 2;

  if (n4 > 0) {
    const int tile = BLOCK * V4_PER_THREAD;   // float4s per block
    const int blocks = (n4 + tile - 1) / tile;
    qhead_main<<<blocks, BLOCK, 0, stream>>>(
        reinterpret_cast<const v4f*>(x), w,
        reinterpret_cast<v4f*>(out), n4);
  }
  const int rem = n & 3;
  if (rem) {
    qhead_tail<<<1, 64, 0, stream>>>(x, w, out, n4 << 2, n);
  }
}